// WeightedSoftAttentionMessage_36120674959713
// MI455X (gfx1250) — compile-verified
//
#include <hip/hip_runtime.h>
#include <hip/hip_bf16.h>

typedef __attribute__((ext_vector_type(2))) float v2f;
typedef __attribute__((ext_vector_type(8))) float v8f;

#define BB   32
#define NN   64
#define DD   64
#define HH   4
#define HID  256
#define K1   128          // 2*D
#define ASTR 132          // padded LDS row stride (words) for the A tile
#define NWAVES 8

__global__ __launch_bounds__(256)
void wsam_kernel(const float* __restrict__ element,
                 const float* __restrict__ fraction,
                 const int*   __restrict__ eidx,
                 const float* __restrict__ sa_w1, const float* __restrict__ sa_b1,
                 const float* __restrict__ sa_w2, const float* __restrict__ sa_b2,
                 const float* __restrict__ ac_w1, const float* __restrict__ ac_b1,
                 const float* __restrict__ ac_w2, const float* __restrict__ ac_b2,
                 float* __restrict__ out)
{
    __shared__ float A[NN * ASTR];     // 64 x 128 concat rows, padded
    __shared__ float awacc[NN];
    __shared__ float wnorm[NN];
    __shared__ float g[HID];
    __shared__ float frac_s[NN];
    __shared__ float invS;

    const int tid  = threadIdx.x;
    const int lane = tid & 31;
    const int wave = tid >> 5;
    const int half = lane >> 4;        // K-pair / M-half select (wave32 WMMA layout)
    const int l16  = lane & 15;

    const int b = blockIdx.x >> 6;
    const int x = blockIdx.x & 63;

    // Per-wave fixed tile geometry: wave owns one M-tile, walks 8 N-tiles.
    const int mt    = wave & 3;                 // M-tile (rows mt*16 .. mt*16+15)
    const int m0    = mt * 16 + l16;            // A-row this lane feeds
    const int nbase = (wave >> 2) * 128 + l16;  // first column this lane touches

    // ---- gather concat rows [64 x 128] into LDS (4 threads per row) ----
    {
        const int y = tid >> 2;
        const int q = tid & 3;
        const int base = b * (2 * NN * NN) + 2 * (x * NN + y);
        const int i0 = eidx[base];
        const int i1 = eidx[base + 1];
        const float4* r0 = (const float4*)(element + ((size_t)b * NN + i0) * DD);
        const float4* r1 = (const float4*)(element + ((size_t)b * NN + i1) * DD);
        float4* dst = (float4*)(A + y * ASTR);
#pragma unroll
        for (int c = 0; c < 8; ++c) {
            const int cc = q + 4 * c;                 // 0..31 float4 chunks
            dst[cc] = (cc < 16) ? r0[cc] : r1[cc - 16];
        }
        if (tid < NN) frac_s[tid] = fraction[b * NN + tid];
    }
    __syncthreads();

    float contrib = 0.0f;   // valid for tid < 64 (d = tid)

    for (int head = 0; head < HH; ++head) {
        if (tid < NN) awacc[tid] = 0.0f;
        g[tid] = 0.0f;
        __syncthreads();

        const float* WS = sa_w1 + (size_t)head * K1 * HID;   // [128][256]
        const float* WC = ac_w1 + (size_t)head * K1 * HID;

        // ---- fused K-pass: one A load feeds 16 WMMAs (8 sa + 8 ac tiles) ----
        v8f accA[8];   // sa accumulators, N-tiles nbase + 16*j
        v8f accB[8];   // ac accumulators
#pragma unroll
        for (int j = 0; j < 8; ++j) { accA[j] = (v8f){}; accB[j] = (v8f){}; }

#pragma unroll 2
        for (int kb = 0; kb < 32; ++kb) {
            const int k0 = kb * 4 + 2 * half;
            const v2f a = *(const v2f*)(A + m0 * ASTR + k0);
            const size_t row0 = (size_t)k0 * HID;
            const size_t row1 = (size_t)(k0 + 1) * HID;
#pragma unroll
            for (int j = 0; j < 8; ++j) {
                const int n = nbase + 16 * j;
                v2f bs, bc;
                bs.x = WS[row0 + n];  bs.y = WS[row1 + n];
                bc.x = WC[row0 + n];  bc.y = WC[row1 + n];
                accA[j] = __builtin_amdgcn_wmma_f32_16x16x4_f32(
                    false, a, false, bs, (short)0, accA[j], false, false);
                accB[j] = __builtin_amdgcn_wmma_f32_16x16x4_f32(
                    false, a, false, bc, (short)0, accB[j], false, false);
            }
        }

        // ---- sa epilogue: bias + leaky + dot(sa_w2), reduced once per wave ----
        {
            float vsum[8];
#pragma unroll
            for (int r = 0; r < 8; ++r) vsum[r] = 0.0f;
#pragma unroll
            for (int j = 0; j < 8; ++j) {
                const int n = nbase + 16 * j;
                const float bias = sa_b1[head * HID + n];
                const float w2v  = sa_w2[head * HID + n];
#pragma unroll
                for (int r = 0; r < 8; ++r) {
                    float v = accA[j][r] + bias;
                    v = fmaxf(v, 0.01f * v);
                    vsum[r] += v * w2v;
                }
            }
#pragma unroll
            for (int r = 0; r < 8; ++r) {
                float v = vsum[r];
                v += __shfl_xor(v, 1);
                v += __shfl_xor(v, 2);
                v += __shfl_xor(v, 4);
                v += __shfl_xor(v, 8);            // stays within 16-lane half
                if (l16 == 0)
                    atomicAdd(&awacc[mt * 16 + r + 8 * half], v);
            }
        }
        __syncthreads();

        if (tid < NN)
            wnorm[tid] = __expf(awacc[tid] + sa_b2[head]) * frac_s[tid];
        __syncthreads();
        if (tid == 0) {
            float s = 0.0f;
            for (int y = 0; y < NN; ++y) s += wnorm[y];
            invS = 1.0f / s;
        }
        __syncthreads();
        if (tid < NN) wnorm[tid] *= invS;
        __syncthreads();

        // ---- ac epilogue: bias + leaky + y-weighted reduce into g[k] ----
        {
            float wn[8];
#pragma unroll
            for (int r = 0; r < 8; ++r)
                wn[r] = wnorm[mt * 16 + r + 8 * half];
#pragma unroll
            for (int j = 0; j < 8; ++j) {
                const int n = nbase + 16 * j;
                const float bias = ac_b1[head * HID + n];
                float s = 0.0f;
#pragma unroll
                for (int r = 0; r < 8; ++r) {
                    float v = accB[j][r] + bias;
                    v = fmaxf(v, 0.01f * v);
                    s += wn[r] * v;
                }
                atomicAdd(&g[n], s);              // both halves hit same n (2-way)
            }
        }
        __syncthreads();

        // ---- tiny GEMM2: contrib[d] += g . ac_w2[:, d] + ac_b2[d] ----
        if (tid < DD) {
            const float* W2 = ac_w2 + (size_t)head * HID * DD;
            float s = ac_b2[head * DD + tid];
#pragma unroll 4
            for (int k = 0; k < HID; ++k)
                s += g[k] * W2[(size_t)k * DD + tid];
            contrib += s;
        }
        __syncthreads();                          // protect g/wnorm before next head
    }

    if (tid < DD) {
        const size_t o = ((size_t)b * NN + x) * DD + tid;
        out[o] = element[o] + contrib * 0.25f;    // mean over 4 heads + residual
    }
}

extern "C" void kernel_launch(void* const* d_in, const int* in_sizes, int n_in,
                              void* d_out, int out_size, void* d_ws, size_t ws_size,
                              hipStream_t stream) {
    const float* element  = (const float*)d_in[0];
    const float* fraction = (const float*)d_in[1];
    const int*   eidx     = (const int*)  d_in[2];
    const float* sa_w1    = (const float*)d_in[3];
    const float* sa_b1    = (const float*)d_in[4];
    const float* sa_w2    = (const float*)d_in[5];
    const float* sa_b2    = (const float*)d_in[6];
    const float* ac_w1    = (const float*)d_in[7];
    const float* ac_b1    = (const float*)d_in[8];
    const float* ac_w2    = (const float*)d_in[9];
    const float* ac_b2    = (const float*)d_in[10];
    float* out = (float*)d_out;

    dim3 grid(BB * NN);   // one workgroup per (b, x)
    dim3 block(256);
    wsam_kernel<<<grid, block, 0, stream>>>(element, fraction, eidx,
                                            sa_w1, sa_b1, sa_w2, sa_b2,
                                            ac_w1, ac_b1, ac_w2, ac_b2, out);
}